// GCN_56745107915453
// MI455X (gfx1250) — compile-verified
//
#include <hip/hip_runtime.h>
#include <hip/hip_bf16.h>

#define N_NODES  50000
#define N_EDGES  640000
#define N_FEAT   128
#define N_HID    128
#define N_CLASS  16
#define N_GRAPHS 64
#define M_TILES  (N_NODES / 16)   // 3125

typedef float v2f __attribute__((ext_vector_type(2)));
typedef float v8f __attribute__((ext_vector_type(8)));

// ---------------------------------------------------------------------------
// degree / norm precompute
// ---------------------------------------------------------------------------
__global__ void k_deg_init(float* __restrict__ deg, int n) {
    int i = blockIdx.x * blockDim.x + threadIdx.x;
    if (i < n) deg[i] = 1.0f;               // self loop contributes 1
}

__global__ void k_deg_edges(const int* __restrict__ cols, float* __restrict__ deg, int nE) {
    int e = blockIdx.x * blockDim.x + threadIdx.x;
    if (e < nE) atomicAdd(&deg[cols[e]], 1.0f);
}

__global__ void k_dinv(float* __restrict__ deg, int n) {
    int i = blockIdx.x * blockDim.x + threadIdx.x;
    if (i < n) {
        float d = deg[i];
        deg[i] = (d > 0.0f) ? rsqrtf(d) : 0.0f;   // in-place: deg -> dinv
    }
}

__global__ void k_enorm(const int* __restrict__ rows, const int* __restrict__ cols,
                        const float* __restrict__ dinv, float* __restrict__ enorm, int nE) {
    int e = blockIdx.x * blockDim.x + threadIdx.x;
    if (e < nE) enorm[e] = dinv[rows[e]] * dinv[cols[e]];
}

// ---------------------------------------------------------------------------
// W transpose: Wt[c][k] = W[k][c]  (so B fragments load as contiguous b64)
// ---------------------------------------------------------------------------
__global__ void k_transposeW(const float* __restrict__ W, float* __restrict__ Wt) {
    int i = blockIdx.x * blockDim.x + threadIdx.x;   // 16384
    if (i >= N_HID * N_HID) return;
    int k = i >> 7, c = i & 127;
    Wt[(size_t)c * 128 + k] = W[i];
}

// ---------------------------------------------------------------------------
// GEMM: Y[Nx128] = X[Nx128] @ W[128x128] via V_WMMA_F32_16X16X4_F32.
// Each wave computes a 16x64 output strip (4 accumulators), reusing one A
// fragment per k-step across 4 B fragments. Block = 8 waves = 4 M-tiles.
// A frag (16x4 f32): lanes 0-15 -> K=kk,kk+1 (kk=k) ; lanes 16-31 -> K=k+2,k+3
// B frag (4x16 f32): same K split, N = lane&15
// C/D (16x16 f32): VGPR r, lanes 0-15 -> M=r ; lanes 16-31 -> M=r+8
// ---------------------------------------------------------------------------
__global__ __launch_bounds__(256)
void k_gemm128_wmma(const float* __restrict__ X, const float* __restrict__ Wt,
                    float* __restrict__ Y) {
    const int wave = threadIdx.x >> 5;
    const int lane = threadIdx.x & 31;
    const int mt   = blockIdx.x * 4 + (wave >> 1);   // wave-uniform
    if (mt >= M_TILES) return;                       // uniform exit: EXEC stays full
    const int l15     = lane & 15;
    const int row     = mt * 16 + l15;
    const int colBase = (wave & 1) * 64;             // 2 waves cover 128 cols
    const int kh      = (lane >> 4) << 1;            // 0 | 2

    const float* Arow = X  + (size_t)row * 128 + kh;
    const float* B0   = Wt + (size_t)(colBase +  0 + l15) * 128 + kh;
    const float* B1   = Wt + (size_t)(colBase + 16 + l15) * 128 + kh;
    const float* B2   = Wt + (size_t)(colBase + 32 + l15) * 128 + kh;
    const float* B3   = Wt + (size_t)(colBase + 48 + l15) * 128 + kh;

    v8f acc0 = {}, acc1 = {}, acc2 = {}, acc3 = {};
    #pragma unroll
    for (int k = 0; k < 128; k += 4) {
        v2f a  = *(const v2f*)(Arow + k);
        v2f b0 = *(const v2f*)(B0 + k);
        v2f b1 = *(const v2f*)(B1 + k);
        v2f b2 = *(const v2f*)(B2 + k);
        v2f b3 = *(const v2f*)(B3 + k);
        acc0 = __builtin_amdgcn_wmma_f32_16x16x4_f32(false, a, false, b0, (short)0, acc0, false, false);
        acc1 = __builtin_amdgcn_wmma_f32_16x16x4_f32(false, a, false, b1, (short)0, acc1, false, false);
        acc2 = __builtin_amdgcn_wmma_f32_16x16x4_f32(false, a, false, b2, (short)0, acc2, false, false);
        acc3 = __builtin_amdgcn_wmma_f32_16x16x4_f32(false, a, false, b3, (short)0, acc3, false, false);
    }

    const int mOff = (lane < 16) ? 0 : 8;
    float* Yb = Y + (size_t)(mt * 16 + mOff) * 128 + colBase + l15;
    #pragma unroll
    for (int r = 0; r < 8; ++r) {
        Yb[(size_t)r * 128 +  0] = acc0[r];
        Yb[(size_t)r * 128 + 16] = acc1[r];
        Yb[(size_t)r * 128 + 32] = acc2[r];
        Yb[(size_t)r * 128 + 48] = acc3[r];
    }
}

// ---------------------------------------------------------------------------
// aggregation: hB = bias ; hB[col] += enorm * hA[row] (atomics) ;
//              hB[i] += dinv[i]^2 * hA[i] ; optional ReLU
// One wave per node/edge -> scalar loads for per-entity metadata.
// ---------------------------------------------------------------------------
__global__ __launch_bounds__(256)
void k_bias_init(float4* __restrict__ hB, const float* __restrict__ bias, int n) {
    int i = blockIdx.x * 8 + (threadIdx.x >> 5);     // wave-uniform node id
    if (i >= n) return;
    int c = threadIdx.x & 31;
    hB[(size_t)i * 32 + c] = ((const float4*)bias)[c];
}

__global__ __launch_bounds__(256)
void k_edge_scatter(const float4* __restrict__ hA, const float* __restrict__ enorm,
                    const int* __restrict__ rows, const int* __restrict__ cols,
                    float* __restrict__ hB, int nE) {
    int e = blockIdx.x * 8 + (threadIdx.x >> 5);     // wave-uniform edge id
    if (e >= nE) return;
    int c = threadIdx.x & 31;
    float nrm = enorm[e];
    int r = rows[e], d = cols[e];
    float4 m = hA[(size_t)r * 32 + c];
    float* dst = hB + (size_t)d * 128 + c * 4;
    atomicAdd(dst + 0, nrm * m.x);
    atomicAdd(dst + 1, nrm * m.y);
    atomicAdd(dst + 2, nrm * m.z);
    atomicAdd(dst + 3, nrm * m.w);
}

__global__ __launch_bounds__(256)
void k_finish(const float4* __restrict__ hA, float4* __restrict__ hB,
              const float* __restrict__ dinv, int n, int doRelu) {
    int i = blockIdx.x * 8 + (threadIdx.x >> 5);     // wave-uniform node id
    if (i >= n) return;
    int c = threadIdx.x & 31;
    float dv = dinv[i];
    float s = dv * dv;                  // self-loop norm
    float4 a = hA[(size_t)i * 32 + c];
    float4 v = hB[(size_t)i * 32 + c];
    v.x += s * a.x; v.y += s * a.y; v.z += s * a.z; v.w += s * a.w;
    if (doRelu) {
        v.x = fmaxf(v.x, 0.0f); v.y = fmaxf(v.y, 0.0f);
        v.z = fmaxf(v.z, 0.0f); v.w = fmaxf(v.w, 0.0f);
    }
    hB[(size_t)i * 32 + c] = v;
}

// ---------------------------------------------------------------------------
// pooling + head
// ---------------------------------------------------------------------------
__global__ void k_pool_init(float* __restrict__ pooled, float* __restrict__ cnt) {
    int i = blockIdx.x * blockDim.x + threadIdx.x;
    if (i < N_GRAPHS * N_HID) pooled[i] = 0.0f;
    if (i < N_GRAPHS) cnt[i] = 0.0f;
}

__global__ __launch_bounds__(256)
void k_pool_sum(const float4* __restrict__ hB, const int* __restrict__ batch,
                float* __restrict__ pooled, float* __restrict__ cnt, int n) {
    int i = blockIdx.x * 8 + (threadIdx.x >> 5);     // wave-uniform node id
    if (i >= n) return;
    int c = threadIdx.x & 31;
    int g = batch[i];
    float4 v = hB[(size_t)i * 32 + c];
    float* dst = pooled + (size_t)g * 128 + c * 4;
    atomicAdd(dst + 0, v.x);
    atomicAdd(dst + 1, v.y);
    atomicAdd(dst + 2, v.z);
    atomicAdd(dst + 3, v.w);
    if (c == 0) atomicAdd(&cnt[g], 1.0f);
}

__global__ void k_head(const float* __restrict__ pooled, const float* __restrict__ cnt,
                       const float* __restrict__ Wlin, const float* __restrict__ blin,
                       float* __restrict__ out) {
    int g = blockIdx.x;          // graph
    int t = threadIdx.x;         // class, 0..15
    __shared__ float sh[N_CLASS];
    float inv = 1.0f / fmaxf(cnt[g], 1.0f);
    float acc = blin[t];
    #pragma unroll 4
    for (int k = 0; k < N_HID; ++k)
        acc += (pooled[(size_t)g * N_HID + k] * inv) * Wlin[(size_t)k * N_CLASS + t];
    sh[t] = acc;
    __syncthreads();
    float m = sh[0];
    #pragma unroll
    for (int i = 1; i < N_CLASS; ++i) m = fmaxf(m, sh[i]);
    float ex = expf(acc - m);
    __syncthreads();
    sh[t] = ex;
    __syncthreads();
    float s = 0.0f;
    #pragma unroll
    for (int i = 0; i < N_CLASS; ++i) s += sh[i];
    out[(size_t)g * N_CLASS + t] = ex / s;
}

// ---------------------------------------------------------------------------
// orchestration
// ---------------------------------------------------------------------------
extern "C" void kernel_launch(void* const* d_in, const int* in_sizes, int n_in,
                              void* d_out, int out_size, void* d_ws, size_t ws_size,
                              hipStream_t stream) {
    const float* x     = (const float*)d_in[0];
    const int*   ei    = (const int*)d_in[1];
    const int*   batch = (const int*)d_in[2];
    const float* W[4]  = {(const float*)d_in[3], (const float*)d_in[5],
                          (const float*)d_in[7], (const float*)d_in[9]};
    const float* b[4]  = {(const float*)d_in[4], (const float*)d_in[6],
                          (const float*)d_in[8], (const float*)d_in[10]};
    const float* Wlin  = (const float*)d_in[11];
    const float* blin  = (const float*)d_in[12];

    const int* rows = ei;             // edge_index[0] = source
    const int* cols = ei + N_EDGES;   // edge_index[1] = target

    float* ws     = (float*)d_ws;
    float* hA     = ws;                                  // [N,128] gemm out
    float* hB     = hA + (size_t)N_NODES * N_HID;        // [N,128] aggregated
    float* dinv   = hB + (size_t)N_NODES * N_HID;        // [N]
    float* enorm  = dinv + N_NODES;                      // [E]
    float* pooled = enorm + N_EDGES;                     // [64,128]
    float* cnt    = pooled + N_GRAPHS * N_HID;           // [64]
    float* Wt     = cnt + N_GRAPHS;                      // [128,128] W^T

    const int B = 256;
    const int gN   = (N_NODES + B - 1) / B;              // per-node
    const int gE   = (N_EDGES + B - 1) / B;              // per-edge
    const int gNw  = (N_NODES + 7) / 8;                  // one wave per node
    const int gEw  = (N_EDGES + 7) / 8;                  // one wave per edge
    const int gMM  = (M_TILES + 3) / 4;                  // 4 M-tiles per block

    // normalization
    k_deg_init<<<gN, B, 0, stream>>>(dinv, N_NODES);
    k_deg_edges<<<gE, B, 0, stream>>>(cols, dinv, N_EDGES);
    k_dinv<<<gN, B, 0, stream>>>(dinv, N_NODES);
    k_enorm<<<gE, B, 0, stream>>>(rows, cols, dinv, enorm, N_EDGES);

    // 4 GCN layers (ReLU on first 3)
    const float* cur = x;
    for (int l = 0; l < 4; ++l) {
        k_transposeW<<<(N_HID * N_HID + B - 1) / B, B, 0, stream>>>(W[l], Wt);
        k_gemm128_wmma<<<gMM, 256, 0, stream>>>(cur, Wt, hA);
        k_bias_init<<<gNw, B, 0, stream>>>((float4*)hB, b[l], N_NODES);
        k_edge_scatter<<<gEw, B, 0, stream>>>((const float4*)hA, enorm, rows, cols,
                                              hB, N_EDGES);
        k_finish<<<gNw, B, 0, stream>>>((const float4*)hA, (float4*)hB, dinv,
                                        N_NODES, (l < 3) ? 1 : 0);
        cur = hB;
    }

    // global mean pool + linear + softmax
    k_pool_init<<<(N_GRAPHS * N_HID + B - 1) / B, B, 0, stream>>>(pooled, cnt);
    k_pool_sum<<<gNw, B, 0, stream>>>((const float4*)hB, batch, pooled, cnt, N_NODES);
    k_head<<<N_GRAPHS, N_CLASS, 0, stream>>>(pooled, cnt, Wlin, blin, (float*)d_out);
}